// Basis_44805098832284
// MI455X (gfx1250) — compile-verified
//
#include <hip/hip_runtime.h>

// out[N,256] = prim[N,1024] x S[1024,256], S one-hot from sorted orbital_index.
// Reduction done with V_WMMA_F32_16X16X4_F32 (exact f32 MAC); per-primitive
// Gaussian evaluation on VALU (v_exp_f32), params staged once in LDS.
typedef __attribute__((ext_vector_type(2))) float v2f;
typedef __attribute__((ext_vector_type(8))) float v8f;

constexpr int N_PRIM = 1024;
constexpr int N_ORB  = 256;
constexpr int ORB_TILES = N_ORB / 16;     // 16 output column tiles
constexpr int POINTS_PER_BLOCK = 128;     // 8 waves x 16 rows
constexpr int THREADS = 256;              // 8 wave32

__global__ __launch_bounds__(THREADS) void basis_wmma_kernel(
    const float* __restrict__ pos,        // [N,3]
    const float* __restrict__ coef,       // [P]
    const float* __restrict__ nrm,        // [P]
    const float* __restrict__ center,     // [P,3]
    const float* __restrict__ alpha,      // [P]
    const int*  __restrict__ lmn,         // [P,3] in {0,1,2}
    const int*  __restrict__ orbidx,      // [P] sorted, in [0,256)
    float* __restrict__ out)              // [N,256]
{
    // 32 KB packed primitive table: {cx,cy,cz,-alpha*log2e},{c*n,lx,ly,(orb<<2)|lz}
    __shared__ float4 s_prim[N_PRIM * 2];
    __shared__ int    s_tile[ORB_TILES + 1];

    const int tid = threadIdx.x;

    // ---- one-time stage of all primitive params into LDS ----
    #pragma unroll
    for (int k = 0; k < N_PRIM / THREADS; ++k) {
        const int p = tid * (N_PRIM / THREADS) + k;
        float4 g;
        g.x = center[p * 3 + 0];
        g.y = center[p * 3 + 1];
        g.z = center[p * 3 + 2];
        g.w = -alpha[p] * 1.4426950408889634f;     // exp(-a*r2) = exp2(g.w*r2)
        float4 h;
        h.x = coef[p] * nrm[p];
        h.y = (float)lmn[p * 3 + 0];
        h.z = (float)lmn[p * 3 + 1];
        h.w = __int_as_float((orbidx[p] << 2) | lmn[p * 3 + 2]); // free decode
        s_prim[p * 2 + 0] = g;
        s_prim[p * 2 + 1] = h;
    }
    // Sorted orbital_index => contiguous primitive range per 16-orbital tile.
    if (tid <= ORB_TILES) {
        const int target = tid * 16;
        int lo = 0, hi = N_PRIM;
        while (lo < hi) {
            const int mid = (lo + hi) >> 1;
            if (orbidx[mid] < target) lo = mid + 1; else hi = mid;
        }
        s_tile[tid] = lo;
    }
    __syncthreads();

    const int lane = tid & 31;
    const int li   = lane & 15;   // A: row M / B: col N owned by this lane
    const int half = lane >> 4;   // K pair: lanes 0-15 hold K={0,1}, 16-31 K={2,3}
    const int wave = tid >> 5;
    const int pbase = blockIdx.x * POINTS_PER_BLOCK + wave * 16;

    const float px = pos[(pbase + li) * 3 + 0];
    const float py = pos[(pbase + li) * 3 + 1];
    const float pz = pos[(pbase + li) * 3 + 2];

    for (int t = 0; t < ORB_TILES; ++t) {
        const int ps  = s_tile[t];       // uniform across wave (LDS broadcast)
        const int pe  = s_tile[t + 1];
        const int tli = t * 16 + li;

        // Evaluate one K=4 chunk starting at p: A values + one-hot B column.
        // Branchless: overhang past pe gets B=0 (loads clamped so A is finite).
        auto eval_chunk = [&](int p, v2f& a, v2f& b) {
            #pragma unroll
            for (int kk = 0; kk < 2; ++kk) {
                const int pk = p + half * 2 + kk;
                const int pc = pk < N_PRIM - 1 ? pk : N_PRIM - 1;
                const float4 g = s_prim[pc * 2 + 0];
                const float4 h = s_prim[pc * 2 + 1];
                const float dx = px - g.x, dy = py - g.y, dz = pz - g.z;
                float r2 = dx * dx;
                r2 = fmaf(dy, dy, r2);
                r2 = fmaf(dz, dz, r2);
                const int enc  = __float_as_int(h.w);
                const int lz   = enc & 3;
                const int orbv = enc >> 2;
                const float mx = (h.y >= 1.0f ? dx : 1.0f) * (h.y >= 2.0f ? dx : 1.0f);
                const float my = (h.z >= 1.0f ? dy : 1.0f) * (h.z >= 2.0f ? dy : 1.0f);
                const float mz = (lz >= 1 ? dz : 1.0f) * (lz >= 2 ? dz : 1.0f);
                const float ex = __builtin_amdgcn_exp2f(g.w * r2); // v_exp_f32
                a[kk] = h.x * mx * my * mz * ex;
                b[kk] = ((pk < pe) & (orbv == tli)) ? 1.0f : 0.0f; // no branch
            }
        };

        // Two independent accumulator chains so v_wmma ops interleave.
        v8f acc0 = {}, acc1 = {};
        for (int p = ps; p < pe; p += 8) {       // uniform loop -> EXEC all ones
            v2f a0, b0, a1, b1;
            eval_chunk(p,     a0, b0);
            eval_chunk(p + 4, a1, b1);
            acc0 = __builtin_amdgcn_wmma_f32_16x16x4_f32(
                false, a0, false, b0, (short)0, acc0, false, false);
            acc1 = __builtin_amdgcn_wmma_f32_16x16x4_f32(
                false, a1, false, b1, (short)0, acc1, false, false);
        }
        const v8f acc = acc0 + acc1;

        // C/D layout: VGPR r -> row (r + 8*half), col (lane&15); 64B row segments
        float* o = out + (size_t)(pbase + half * 8) * N_ORB + t * 16 + li;
        #pragma unroll
        for (int r = 0; r < 8; ++r) {
            o[(size_t)r * N_ORB] = acc[r];
        }
    }
}

extern "C" void kernel_launch(void* const* d_in, const int* in_sizes, int n_in,
                              void* d_out, int out_size, void* d_ws, size_t ws_size,
                              hipStream_t stream) {
    const float* pos    = (const float*)d_in[0];
    const float* coef   = (const float*)d_in[1];
    const float* nrm    = (const float*)d_in[2];
    const float* center = (const float*)d_in[3];
    const float* alpha  = (const float*)d_in[4];
    const int*   lmn    = (const int*)d_in[5];
    const int*   orbidx = (const int*)d_in[6];
    float*       out    = (float*)d_out;

    const int n_points = in_sizes[0] / 3;            // 65536
    const int grid = n_points / POINTS_PER_BLOCK;    // 512 blocks of 8 waves

    basis_wmma_kernel<<<grid, THREADS, 0, stream>>>(
        pos, coef, nrm, center, alpha, lmn, orbidx, out);
}